// Transformer_61529701483238
// MI455X (gfx1250) — compile-verified
//
#include <hip/hip_runtime.h>
#include <hip/hip_bf16.h>

// ---------------------------------------------------------------------------
// Problem constants (match reference)
// ---------------------------------------------------------------------------
#define L_LAYERS 6
#define BS 2
#define HH 48
#define WW 192
#define CCH 128
#define NHEAD 8
#define DIMH 16
#define NTOK (BS * HH * WW)        // 18432 tokens
#define FEAT (NTOK * CCH)          // 2359296 floats per feature buffer
#define SEQ (HH * WW)              // 9216 tokens per batch (self attention)
#define NROWS (BS * HH)            // 96 cross-attention rows
#define ATTN_SZ (NROWS * WW * WW)  // 3538944 floats per attn accumulator

// LDS strides (chosen so b64 WMMA-operand reads are bank-conflict-free:
// pair-plane strides are == 16 mod 32 -> lane groups hit disjoint bank-pairs)
#define WPSTR 80                   // gemm: W pair-plane stride (float2 units)
#define QPSTR 208                  // cross: Qt/Kt pair-plane stride (float2 units)
#define SSTR 200                   // cross: S row stride (float units)

typedef __attribute__((ext_vector_type(2))) float v2f;
typedef __attribute__((ext_vector_type(8))) float v8f;

// D = A(16x4 f32) x B(4x16 f32) + C(16x16 f32), full fp32 (matches reference).
#define WMMA_F32X4(a, b, c) \
  __builtin_amdgcn_wmma_f32_16x16x4_f32(false, (a), false, (b), (short)0, (c), false, false)

__device__ __forceinline__ v8f zero8() {
  v8f z;
#pragma unroll
  for (int i = 0; i < 8; ++i) z[i] = 0.0f;
  return z;
}

// phi(x) = elu(x) + 1  ->  x>0 ? x+1 : exp(x)
__device__ __forceinline__ float elu1(float x) { return x > 0.0f ? x + 1.0f : __expf(x); }
// exact GELU: 0.5 x (1 + erf(x/sqrt(2)))
__device__ __forceinline__ float gelu_exact(float x) {
  return 0.5f * x * (1.0f + erff(x * 0.70710678118654752f));
}

// ---------------------------------------------------------------------------
// LayerNorm over C=128: one wave32 per row, 4 elems per lane.
// ---------------------------------------------------------------------------
__global__ __launch_bounds__(256) void ln_kernel(const float* __restrict__ X,
                                                 const float* __restrict__ G,
                                                 const float* __restrict__ B,
                                                 float* __restrict__ Y) {
  int lane = threadIdx.x & 31;
  int wave = threadIdx.x >> 5;
  int row = blockIdx.x * 8 + wave;
  const float4 x = ((const float4*)(X + (size_t)row * CCH))[lane];
  float s = x.x + x.y + x.z + x.w;
#pragma unroll
  for (int o = 16; o >= 1; o >>= 1) s += __shfl_xor(s, o, 32);
  float mean = s * (1.0f / 128.0f);
  float dx = x.x - mean, dy = x.y - mean, dz = x.z - mean, dw = x.w - mean;
  float ss = dx * dx + dy * dy + dz * dz + dw * dw;
#pragma unroll
  for (int o = 16; o >= 1; o >>= 1) ss += __shfl_xor(ss, o, 32);
  float rstd = rsqrtf(ss * (1.0f / 128.0f) + 1e-5f);
  const float4 g = ((const float4*)G)[lane];
  const float4 b = ((const float4*)B)[lane];
  float4 y;
  y.x = dx * rstd * g.x + b.x;
  y.y = dy * rstd * g.y + b.y;
  y.z = dz * rstd * g.z + b.z;
  y.w = dw * rstd * g.w + b.w;
  ((float4*)(Y + (size_t)row * CCH))[lane] = y;
}

// ---------------------------------------------------------------------------
// GEMM: Y[n,o] = act( sum_i X[n,i] * W[o,i] ) + R[n,o]
// Block (8 waves) stages W in LDS as k-PAIRED float2 planes:
//   Wp[p][o] = {W[o][2p], W[o][2p+1]}   (p = k/2, plane stride WPSTR)
// A lane's WMMA B operand (k = kb+kh, kb+kh+1; kh even) is then ONE aligned
// ds_load_b64 -> no register-pair shuffles, conflict-free banks.
// A streams from global (contiguous b64 per lane). 8 WMMA accumulators/wave.
// grid.x = NTOK/128 = 144, dynamic LDS = 64*WPSTR*8 = 40960 B.
// ---------------------------------------------------------------------------
__global__ __launch_bounds__(256) void gemm_kernel(const float* __restrict__ X,
                                                   const float* __restrict__ W,
                                                   const float* __restrict__ R,
                                                   float* __restrict__ Y, int act) {
  extern __shared__ v2f Wp[];  // [64 pair-planes][WPSTR]
  int lane = threadIdx.x & 31;
  int wave = threadIdx.x >> 5;
  int m = lane & 15;             // row within tile (A) / col within tile (B,D)
  int kh = (lane < 16) ? 0 : 2;  // K sub-offset per ISA A/B f32 layout
  int rowBase = blockIdx.x * 128 + wave * 16;

  // Stage W -> LDS paired-transposed. LDS writes: consecutive o -> no conflicts.
  for (int e = threadIdx.x; e < 128 * 32; e += 256) {
    int o = e & 127;
    int q = e >> 7;  // k quad index, k4 = 4q
    const float4 w = *(const float4*)(W + (size_t)o * CCH + 4 * q);
    v2f lo, hi;
    lo.x = w.x; lo.y = w.y;
    hi.x = w.z; hi.y = w.w;
    Wp[(2 * q + 0) * WPSTR + o] = lo;
    Wp[(2 * q + 1) * WPSTR + o] = hi;
  }
  __syncthreads();

  v8f acc[8];
#pragma unroll
  for (int t = 0; t < 8; ++t) acc[t] = zero8();

  const float* xrow = X + (size_t)(rowBase + m) * CCH;
  for (int kb = 0; kb < 128; kb += 4) {
    v2f a = *(const v2f*)(xrow + kb + kh);       // one global b64
    const v2f* wrow = Wp + ((kb + kh) >> 1) * WPSTR + m;
#pragma unroll
    for (int t = 0; t < 8; ++t) {
      v2f b = wrow[t * 16];                      // one ds_load_b64
      acc[t] = WMMA_F32X4(a, b, acc[t]);
    }
  }

#pragma unroll
  for (int t = 0; t < 8; ++t) {
    int col = t * 16 + m;
#pragma unroll
    for (int r = 0; r < 8; ++r) {
      int row = rowBase + r + ((lane < 16) ? 0 : 8);
      float v = acc[t][r];
      if (act) v = gelu_exact(v);
      if (R) v += R[(size_t)row * CCH + col];
      Y[(size_t)row * CCH + col] = v;
    }
  }
}

// ---------------------------------------------------------------------------
// Linear attention stage 1: per (batch, head) compute
//   KV[d,e] = sum_s phi(K[s,d]) * V[s,e]   (16x16, WMMA reduction over S=9216)
//   Ksum[d] = sum_s phi(K[s,d])
// KV is stored pair-interleaved: KV[(d/2)*32 + e*2 + (d&1)] so that stage 2's
// B operand is one contiguous 8B load per lane.
// grid = 16 blocks (b*8+h), 256 threads (8 waves split the S range).
// ---------------------------------------------------------------------------
__global__ __launch_bounds__(256) void kv_kernel(const float* __restrict__ K,
                                                 const float* __restrict__ V,
                                                 float* __restrict__ KV,
                                                 float* __restrict__ Ksum) {
  __shared__ float red[8 * 256];
  __shared__ float ksred[256];
  int lane = threadIdx.x & 31;
  int wave = threadIdx.x >> 5;
  int bh = blockIdx.x;
  int b = bh >> 3, hh = bh & 7;
  int ch = hh * DIMH;
  size_t tok0 = (size_t)b * SEQ;
  int m = lane & 15;
  int kh = (lane < 16) ? 0 : 2;

  v8f acc = zero8();
  int s0 = wave * (SEQ / 8), s1 = s0 + (SEQ / 8);
  for (int s = s0; s < s1; s += 4) {
    size_t base = (tok0 + s + kh) * CCH + ch;
    v2f a;  // A[d, j] = phi(K[s+j, d]) ; d = m
    a.x = elu1(K[base + m]);
    a.y = elu1(K[base + CCH + m]);
    v2f bb;  // B[j, e] = V[s+j, e] ; e = m
    bb.x = V[base + m];
    bb.y = V[base + CCH + m];
    acc = WMMA_F32X4(a, bb, acc);
  }
#pragma unroll
  for (int r = 0; r < 8; ++r)
    red[wave * 256 + (r + ((lane < 16) ? 0 : 8)) * 16 + m] = acc[r];

  // Ksum partials: thread t handles d = t%16 over a 576-token chunk.
  int d = threadIdx.x & 15, chunk = threadIdx.x >> 4;
  float sum = 0.0f;
  for (int s = chunk * 576; s < chunk * 576 + 576; ++s)
    sum += elu1(K[(tok0 + s) * CCH + ch + d]);
  ksred[threadIdx.x] = sum;
  __syncthreads();

  float a2 = 0.0f;
  for (int w = 0; w < 8; ++w) a2 += red[w * 256 + threadIdx.x];
  {  // pair-interleaved store: t = dd*16 + ee
    int dd = threadIdx.x >> 4, ee = threadIdx.x & 15;
    KV[bh * 256 + (dd >> 1) * 32 + ee * 2 + (dd & 1)] = a2;
  }
  if (threadIdx.x < 16) {
    float ks = 0.0f;
    for (int c = 0; c < 16; ++c) ks += ksred[c * 16 + threadIdx.x];
    Ksum[bh * 16 + threadIdx.x] = ks;
  }
}

// ---------------------------------------------------------------------------
// Linear attention stage 2:
//   out[l,e] = (phi(Q[l,:]) @ KV) / (phi(Q[l,:]) . Ksum + 1e-6)
// One wave per (b, h, 16-row tile). grid = 2*8*576/8 = 1152 blocks.
// ---------------------------------------------------------------------------
__global__ __launch_bounds__(256) void linout_kernel(const float* __restrict__ Q,
                                                     const float* __restrict__ KV,
                                                     const float* __restrict__ Ksum,
                                                     float* __restrict__ M) {
  int lane = threadIdx.x & 31;
  int wave = threadIdx.x >> 5;
  int gw = blockIdx.x * 8 + wave;
  int bh = gw / 576, lt = gw % 576;
  int b = bh >> 3, hh = bh & 7;
  int ch = hh * DIMH;
  size_t tokL = (size_t)b * SEQ + (size_t)lt * 16;
  int m = lane & 15;
  int kh = (lane < 16) ? 0 : 2;

  v8f acc = zero8();
#pragma unroll
  for (int kb = 0; kb < 16; kb += 4) {
    size_t qi = (tokL + m) * CCH + ch + kb + kh;
    v2f a;
    a.x = elu1(Q[qi]);
    a.y = elu1(Q[qi + 1]);
    // pair-interleaved KV: one contiguous 8B load per lane
    v2f bb = *(const v2f*)(KV + bh * 256 + ((kb + kh) >> 1) * 32 + m * 2);
    acc = WMMA_F32X4(a, bb, acc);
  }
#pragma unroll
  for (int r = 0; r < 8; ++r) {
    int row = r + ((lane < 16) ? 0 : 8);
    const float* qr = Q + (tokL + row) * CCH + ch;
    float dot = 1e-6f;
#pragma unroll
    for (int d = 0; d < 16; ++d) dot += elu1(qr[d]) * Ksum[bh * 16 + d];
    M[(tokL + row) * CCH + ch + m] = acc[r] / dot;
  }
}

// ---------------------------------------------------------------------------
// Cross attention: one block per n in [0,96), heads looped inside.
// Per head: stage Qt/Kt k-PAIRED in LDS (Qt[p][l] = {Q[l,2p],Q[l,2p+1]},
// stride QPSTR == 16 mod 32 -> conflict-free single-b64 WMMA operand reads);
// S = scale*QK^T in LDS (row stride SSTR=200, phase-3 A = one aligned v2f
// load, 2-way worst case); softmax per row; raw += A accumulated to global
// (same thread owns (row,s) across heads -> no atomics, deterministic);
// out = A @ V via WMMA.
// Dynamic LDS = 192*200*4 + 2*8*QPSTR*8 = 180224 B (<= 320KB WGP LDS).
// ---------------------------------------------------------------------------
__global__ __launch_bounds__(256) void cross_kernel(const float* __restrict__ Q,
                                                    const float* __restrict__ K,
                                                    const float* __restrict__ V,
                                                    float* __restrict__ M,
                                                    float* __restrict__ raw) {
  extern __shared__ float lds[];
  float* S = lds;                                  // [192][SSTR] floats
  v2f* Qt = (v2f*)(lds + WW * SSTR);               // [8 pair-planes][QPSTR]
  v2f* Kt = Qt + 8 * QPSTR;                        // [8 pair-planes][QPSTR]
  int lane = threadIdx.x & 31;
  int wave = threadIdx.x >> 5;
  int n = blockIdx.x;
  size_t tok0 = (size_t)n * WW;
  float* rawn = raw + (size_t)n * WW * WW;
  int m = lane & 15;
  int kh = (lane < 16) ? 0 : 2;
  const float scale = 0.25f;  // 1/sqrt(16)

  for (int hh = 0; hh < NHEAD; ++hh) {
    int ch = hh * DIMH;

    // Stage per-head Q,K slices, k-paired. Writes: consecutive l -> no
    // conflicts. Reads are 8B global loads (L2-hot activations).
    for (int e = threadIdx.x; e < 8 * WW; e += 256) {
      int p = e / WW;
      int l = e - p * WW;
      Qt[p * QPSTR + l] = *(const v2f*)(Q + (tok0 + l) * CCH + ch + 2 * p);
      Kt[p * QPSTR + l] = *(const v2f*)(K + (tok0 + l) * CCH + ch + 2 * p);
    }
    __syncthreads();

    // Phase 1: S tiles (12x12 tiles of 16x16), K-dim = head dim 16, LDS-fed.
    for (int tile = wave; tile < 144; tile += 8) {
      int tl = tile / 12, ts = tile % 12;
      v8f acc = zero8();
#pragma unroll
      for (int kb = 0; kb < 16; kb += 4) {
        int p0 = (kb + kh) >> 1;
        v2f a = Qt[p0 * QPSTR + tl * 16 + m];   // one ds_load_b64
        v2f bb = Kt[p0 * QPSTR + ts * 16 + m];  // one ds_load_b64
        acc = WMMA_F32X4(a, bb, acc);
      }
#pragma unroll
      for (int r = 0; r < 8; ++r)
        S[(tl * 16 + r + ((lane < 16) ? 0 : 8)) * SSTR + ts * 16 + m] = acc[r] * scale;
    }
    __syncthreads();

    // Phase 2: softmax per row (wave per row) + accumulate raw = sum_h A.
    for (int row = wave; row < WW; row += 8) {
      float* Sr = S + row * SSTR;
      float mx = -3.402823466e38f;
      for (int s = lane; s < WW; s += 32) mx = fmaxf(mx, Sr[s]);
#pragma unroll
      for (int o = 16; o >= 1; o >>= 1) mx = fmaxf(mx, __shfl_xor(mx, o, 32));
      float sum = 0.0f;
      for (int s = lane; s < WW; s += 32) {
        float e = __expf(Sr[s] - mx);
        Sr[s] = e;
        sum += e;
      }
#pragma unroll
      for (int o = 16; o >= 1; o >>= 1) sum += __shfl_xor(sum, o, 32);
      float inv = 1.0f / sum;
      for (int s = lane; s < WW; s += 32) {
        float a = Sr[s] * inv;
        Sr[s] = a;
        rawn[row * WW + s] += a;  // same thread owns (row,s) across heads
      }
    }
    __syncthreads();

    // Phase 3: out = A @ V  (A from LDS as aligned v2f, K-dim = 192).
    for (int tl = wave; tl < 12; tl += 8) {
      v8f acc = zero8();
      for (int sb = 0; sb < WW; sb += 4) {
        v2f a = *(const v2f*)(S + (tl * 16 + m) * SSTR + sb + kh);
        const float* vrow = V + (tok0 + sb + kh) * CCH + ch;
        v2f bb;  // B[j, e] = V[sb+j, e]
        bb.x = vrow[m];
        bb.y = vrow[CCH + m];
        acc = WMMA_F32X4(a, bb, acc);
      }
#pragma unroll
      for (int r = 0; r < 8; ++r)
        M[(tok0 + tl * 16 + r + ((lane < 16) ? 0 : 8)) * CCH + ch + m] = acc[r];
    }
    __syncthreads();  // S/Qt/Kt reused next head
  }
}

// ---------------------------------------------------------------------------
// Host orchestration
// ---------------------------------------------------------------------------
extern "C" void kernel_launch(void* const* d_in, const int* in_sizes, int n_in,
                              void* d_out, int out_size, void* d_ws, size_t ws_size,
                              hipStream_t stream) {
  (void)in_sizes; (void)n_in; (void)out_size; (void)ws_size;

  // setup_inputs() dict order (sa_/ca_ interleaved inside the loops):
  const float* feat_l = (const float*)d_in[0];
  const float* feat_r = (const float*)d_in[1];
  const float* attn_l = (const float*)d_in[2];
  const float* attn_r = (const float*)d_in[3];
  const float* SAq = (const float*)d_in[4];
  const float* CAq = (const float*)d_in[5];
  const float* SAk = (const float*)d_in[6];
  const float* CAk = (const float*)d_in[7];
  const float* SAv = (const float*)d_in[8];
  const float* CAv = (const float*)d_in[9];
  const float* SAmg = (const float*)d_in[10];
  const float* CAmg = (const float*)d_in[11];
  const float* SAm1 = (const float*)d_in[12];
  const float* CAm1 = (const float*)d_in[13];
  const float* SAm2 = (const float*)d_in[14];
  const float* CAm2 = (const float*)d_in[15];
  const float* SAn1g = (const float*)d_in[16];
  const float* CAn1g = (const float*)d_in[17];
  const float* SAn1b = (const float*)d_in[18];
  const float* CAn1b = (const float*)d_in[19];
  const float* SAn2g = (const float*)d_in[20];
  const float* CAn2g = (const float*)d_in[21];
  const float* SAn2b = (const float*)d_in[22];
  const float* CAn2b = (const float*)d_in[23];

  float* ws = (float*)d_ws;
  float* fl = ws + (size_t)0 * FEAT;
  float* fr = ws + (size_t)1 * FEAT;
  float* na = ws + (size_t)2 * FEAT;  // LN output A
  float* nb = ws + (size_t)3 * FEAT;  // LN output B
  float* qb = ws + (size_t)4 * FEAT;
  float* kb = ws + (size_t)5 * FEAT;
  float* vb = ws + (size_t)6 * FEAT;
  float* mA = ws + (size_t)7 * FEAT;
  float* mB = ws + (size_t)8 * FEAT;
  float* hid = ws + (size_t)9 * FEAT;
  float* kvb = ws + (size_t)10 * FEAT;  // 16 * 256 (pair-interleaved)
  float* ksb = kvb + 16 * 256;          // 16 * 16

  float* out = (float*)d_out;
  float* out_fl = out;
  float* out_fr = out + (size_t)FEAT;
  float* out_al = out + (size_t)2 * FEAT;
  float* out_ar = out + (size_t)2 * FEAT + ATTN_SZ;

  // Stage inputs (graph-capture-safe d2d copies). attn accumulators are
  // re-initialized from inputs every call -> deterministic replays.
  hipMemcpyAsync(fl, feat_l, (size_t)FEAT * 4, hipMemcpyDeviceToDevice, stream);
  hipMemcpyAsync(fr, feat_r, (size_t)FEAT * 4, hipMemcpyDeviceToDevice, stream);
  hipMemcpyAsync(out_al, attn_l, (size_t)ATTN_SZ * 4, hipMemcpyDeviceToDevice, stream);
  hipMemcpyAsync(out_ar, attn_r, (size_t)ATTN_SZ * 4, hipMemcpyDeviceToDevice, stream);

  dim3 blk(256);
  dim3 gLN(NTOK / 8);
  dim3 gGEMM(NTOK / 128);
  dim3 gKV(16);
  dim3 gLO(NTOK * NHEAD / 16 / 8);  // 1152
  dim3 gCR(NROWS);
  const size_t gemmLds = (size_t)64 * WPSTR * sizeof(v2f);                       // 40960 B
  const size_t crossLds = ((size_t)WW * SSTR + 2 * 8 * QPSTR * 2) * sizeof(float);  // 180224 B

  for (int l = 0; l < L_LAYERS; ++l) {
    size_t wo = (size_t)l * CCH * CCH;
    size_t vo = (size_t)l * CCH;

    // ---- self layer applied to fl, then fr (same params) ----
    for (int side = 0; side < 2; ++side) {
      float* f = side == 0 ? fl : fr;
      ln_kernel<<<gLN, blk, 0, stream>>>(f, SAn1g + vo, SAn1b + vo, na);
      gemm_kernel<<<gGEMM, blk, gemmLds, stream>>>(na, SAq + wo, nullptr, qb, 0);
      gemm_kernel<<<gGEMM, blk, gemmLds, stream>>>(na, SAk + wo, nullptr, kb, 0);
      gemm_kernel<<<gGEMM, blk, gemmLds, stream>>>(na, SAv + wo, nullptr, vb, 0);
      kv_kernel<<<gKV, blk, 0, stream>>>(kb, vb, kvb, ksb);
      linout_kernel<<<gLO, blk, 0, stream>>>(qb, kvb, ksb, mA);
      gemm_kernel<<<gGEMM, blk, gemmLds, stream>>>(mA, SAmg + wo, f, mB, 0);  // msg = f + merge
      ln_kernel<<<gLN, blk, 0, stream>>>(mB, SAn2g + vo, SAn2b + vo, na);
      gemm_kernel<<<gGEMM, blk, gemmLds, stream>>>(na, SAm1 + wo, nullptr, hid, 1);  // GELU
      gemm_kernel<<<gGEMM, blk, gemmLds, stream>>>(hid, SAm2 + wo, mB, f, 0);  // f = msg + mlp2
    }

    // ---- cross layer ----
    ln_kernel<<<gLN, blk, 0, stream>>>(fl, CAn1g + vo, CAn1b + vo, na);  // fln
    ln_kernel<<<gLN, blk, 0, stream>>>(fr, CAn1g + vo, CAn1b + vo, nb);  // frn

    // half 1: ql with kr, vr -> mA, raw -> out_al
    gemm_kernel<<<gGEMM, blk, gemmLds, stream>>>(na, CAq + wo, nullptr, qb, 0);
    gemm_kernel<<<gGEMM, blk, gemmLds, stream>>>(nb, CAk + wo, nullptr, kb, 0);
    gemm_kernel<<<gGEMM, blk, gemmLds, stream>>>(nb, CAv + wo, nullptr, vb, 0);
    cross_kernel<<<gCR, blk, crossLds, stream>>>(qb, kb, vb, mA, out_al);

    // half 2: qr with kl, vl -> mB, raw -> out_ar
    gemm_kernel<<<gGEMM, blk, gemmLds, stream>>>(nb, CAq + wo, nullptr, qb, 0);
    gemm_kernel<<<gGEMM, blk, gemmLds, stream>>>(na, CAk + wo, nullptr, kb, 0);
    gemm_kernel<<<gGEMM, blk, gemmLds, stream>>>(na, CAv + wo, nullptr, vb, 0);
    cross_kernel<<<gCR, blk, crossLds, stream>>>(qb, kb, vb, mB, out_ar);

    // merge + MLP, left (qb free -> merged buffer)
    gemm_kernel<<<gGEMM, blk, gemmLds, stream>>>(mA, CAmg + wo, fl, qb, 0);
    ln_kernel<<<gLN, blk, 0, stream>>>(qb, CAn2g + vo, CAn2b + vo, na);
    gemm_kernel<<<gGEMM, blk, gemmLds, stream>>>(na, CAm1 + wo, nullptr, hid, 1);
    gemm_kernel<<<gGEMM, blk, gemmLds, stream>>>(hid, CAm2 + wo, qb, fl, 0);

    // merge + MLP, right (kb free -> merged buffer)
    gemm_kernel<<<gGEMM, blk, gemmLds, stream>>>(mB, CAmg + wo, fr, kb, 0);
    ln_kernel<<<gLN, blk, 0, stream>>>(kb, CAn2g + vo, CAn2b + vo, na);
    gemm_kernel<<<gGEMM, blk, gemmLds, stream>>>(na, CAm1 + wo, nullptr, hid, 1);
    gemm_kernel<<<gGEMM, blk, gemmLds, stream>>>(hid, CAm2 + wo, kb, fr, 0);
  }

  hipMemcpyAsync(out_fl, fl, (size_t)FEAT * 4, hipMemcpyDeviceToDevice, stream);
  hipMemcpyAsync(out_fr, fr, (size_t)FEAT * 4, hipMemcpyDeviceToDevice, stream);
}